// CosineSim3D_28157805592887
// MI455X (gfx1250) — compile-verified
//
#include <hip/hip_runtime.h>
#include <hip/hip_bf16.h>
#include <math.h>

#define BATCH   128
#define NROWS   1024
#define DFEAT   300
#define EPSV    1e-7f
#define SPLITB  4

typedef float v8f __attribute__((ext_vector_type(8)));
typedef float v2f __attribute__((ext_vector_type(2)));

#if defined(__has_builtin)
#if __has_builtin(__builtin_amdgcn_wmma_f32_16x16x4_f32)
#define HAVE_WMMA_F32X4 1
#endif
#endif

__device__ __forceinline__ float waveRedSum(float v) {
#pragma unroll
    for (int m = 16; m >= 1; m >>= 1) v += __shfl_xor(v, m, 32);
    return v;
}
__device__ __forceinline__ float waveRedMax(float v) {
#pragma unroll
    for (int m = 16; m >= 1; m >>= 1) v = fmaxf(v, __shfl_xor(v, m, 32));
    return v;
}

// ---------------- K0: zero s_b accumulator region of workspace ----------------
__global__ void k0_zero(float* ws_s, int n) {
    int i = blockIdx.x * blockDim.x + threadIdx.x;
    if (i < n) ws_s[i] = 0.0f;
}

// ---------------- K1: s_b[d] = sum_m b[b,m,d] / |b_m| -------------------------
// grid: BATCH*SPLITB blocks, 256 threads (8 waves). Wave owns 32 rows.
__global__ void k1_bnorm_accum(const float* __restrict__ b, float* __restrict__ ws_s) {
    const int batch = blockIdx.x / SPLITB;
    const int split = blockIdx.x % SPLITB;
    const int wid  = threadIdx.x >> 5;
    const int lane = threadIdx.x & 31;

    float acc[10];
#pragma unroll
    for (int j = 0; j < 10; ++j) acc[j] = 0.0f;

    for (int r = wid; r < 256; r += 8) {
        const int row = split * 256 + r;
        const float* rp = b + ((size_t)batch * NROWS + row) * DFEAT;
        // Prefetch this wave's next row while we crunch the current one.
        if (r + 8 < 256) {
            __builtin_prefetch(rp + 8 * DFEAT + lane * 8, 0, 0);
        }
        float v[10];
        float ss = 0.0f;
#pragma unroll
        for (int j = 0; j < 10; ++j) {
            const int d = lane + 32 * j;
            v[j] = (d < DFEAT) ? rp[d] : 0.0f;
            ss = fmaf(v[j], v[j], ss);
        }
        ss = waveRedSum(ss);
        const float inv = 1.0f / sqrtf(fmaxf(ss, EPSV));
#pragma unroll
        for (int j = 0; j < 10; ++j) acc[j] = fmaf(v[j], inv, acc[j]);
    }

    float* sb = ws_s + (size_t)batch * DFEAT;
#pragma unroll
    for (int j = 0; j < 10; ++j) {
        const int d = lane + 32 * j;
        if (d < DFEAT) unsafeAtomicAdd(&sb[d], acc[j]);
    }
}

// ---------------- K2: scores[b,n] = (a_n . s_b) / |a_n|  via WMMA f32 16x16x4 -
// grid: BATCH*SPLITB blocks, 256 threads (8 waves). Wave owns 16-row tiles.
__global__ void k2_scores_wmma(const float* __restrict__ a,
                               const float* __restrict__ ws_s,
                               float* __restrict__ ws_scores) {
    __shared__ float lds_s[DFEAT];
    const int batch = blockIdx.x / SPLITB;
    const int split = blockIdx.x % SPLITB;
    const int wid  = threadIdx.x >> 5;
    const int lane = threadIdx.x & 31;

    for (int i = threadIdx.x; i < DFEAT; i += 256)
        lds_s[i] = ws_s[(size_t)batch * DFEAT + i];
    __syncthreads();

    const int lane_hi = (lane >= 16) ? 1 : 0;
    const int khalf   = lane_hi * 2;

    // 16 tiles per block of 256 rows, 8 waves -> 2 tiles per wave
    for (int tile = wid; tile < 16; tile += 8) {
        const int n0  = split * 256 + tile * 16;
        const int row = n0 + (lane & 15);
        const float* rp = a + ((size_t)batch * NROWS + row) * DFEAT;

        float ssacc = 0.0f;
#ifdef HAVE_WMMA_F32X4
        v8f c = {};
        // unroll 5 (75 = 15*5): enough loads in flight for latency hiding,
        // small enough that A/B operands stay in VGPRs (no scratch spills).
#pragma unroll 5
        for (int k0 = 0; k0 < DFEAT; k0 += 4) {        // 75 iterations, 300 % 4 == 0
            const int kb = k0 + khalf;
            const float2 av2 = *(const float2*)(rp + kb);   // 8B aligned
            v2f av; av.x = av2.x; av.y = av2.y;
            v2f bv; bv.x = lds_s[kb]; bv.y = lds_s[kb + 1];
            ssacc = fmaf(av.x, av.x, ssacc);
            ssacc = fmaf(av.y, av.y, ssacc);
            // D = A(16x4) * B(4x16, s_b broadcast over N) + C
            c = __builtin_amdgcn_wmma_f32_16x16x4_f32(
                    false, av, false, bv, (short)0, c, false, false);
        }
#else
        float dacc = 0.0f;
#pragma unroll 5
        for (int k0 = 0; k0 < DFEAT; k0 += 4) {
            const int kb = k0 + khalf;
            const float2 av2 = *(const float2*)(rp + kb);
            ssacc = fmaf(av2.x, av2.x, ssacc);
            ssacc = fmaf(av2.y, av2.y, ssacc);
            dacc  = fmaf(av2.x, lds_s[kb],     dacc);
            dacc  = fmaf(av2.y, lds_s[kb + 1], dacc);
        }
#endif
        float ss = ssacc + __shfl_xor(ssacc, 16, 32);   // full sum for row lane&15
        const float inv_a = 1.0f / sqrtf(fmaxf(ss, EPSV));

#ifdef HAVE_WMMA_F32X4
        // D layout: VGPR i -> row i (lanes 0-15, N=lane) / row 8+i (lanes 16-31).
        // B was broadcast over N, so any column holds the dot product.
#pragma unroll
        for (int i = 0; i < 8; ++i) {
            const float invi = __shfl(inv_a, lane_hi * 8 + i, 32); // inv of row (lane_hi*8+i)
            if (lane == 0 || lane == 16) {
                ws_scores[(size_t)batch * NROWS + n0 + lane_hi * 8 + i] = c[i] * invi;
            }
        }
#else
        float dot = dacc + __shfl_xor(dacc, 16, 32);
        if (lane < 16) {
            ws_scores[(size_t)batch * NROWS + n0 + lane] = dot * inv_a;
        }
#endif
    }
}

// ---------------- K3: softmax over n per batch, in place in ws_scores ---------
__global__ void k3_softmax(float* __restrict__ ws_scores) {
    __shared__ float red[32];
    __shared__ float bcast;
    const int b    = blockIdx.x;
    const int tid  = threadIdx.x;
    const int wid  = tid >> 5;
    const int lane = tid & 31;

    const float score = ws_scores[(size_t)b * NROWS + tid];

    // block max
    float m = waveRedMax(score);
    if (lane == 0) red[wid] = m;
    __syncthreads();
    if (wid == 0) {
        float v = red[lane];
        v = waveRedMax(v);
        if (lane == 0) bcast = v;
    }
    __syncthreads();
    const float mx = bcast;

    const float e = expf(score - mx);

    // block sum
    float s = waveRedSum(e);
    __syncthreads();
    if (lane == 0) red[wid] = s;
    __syncthreads();
    if (wid == 0) {
        float v = red[lane];
        v = waveRedSum(v);
        if (lane == 0) bcast = v;
    }
    __syncthreads();
    const float denom = bcast;

    ws_scores[(size_t)b * NROWS + tid] = e / denom;
}

// ---------------- K4: out[b,n,:] = probs[b,n] (tile to 300) -------------------
// grid: BATCH*NROWS/8 blocks, 256 threads (8 waves); one wave per output row.
__global__ void k4_tile_out(const float* __restrict__ ws_scores, float* __restrict__ out) {
    const int wid  = threadIdx.x >> 5;
    const int lane = threadIdx.x & 31;
    const size_t gr = (size_t)blockIdx.x * 8 + wid;   // global row in [0, 131072)
    const float p = ws_scores[gr];
    float* rp = out + gr * DFEAT;
#pragma unroll
    for (int j = 0; j < 10; ++j) {
        const int d = lane + 32 * j;
        if (d < DFEAT) rp[d] = p;
    }
}

extern "C" void kernel_launch(void* const* d_in, const int* in_sizes, int n_in,
                              void* d_out, int out_size, void* d_ws, size_t ws_size,
                              hipStream_t stream) {
    const float* a = (const float*)d_in[0];
    const float* b = (const float*)d_in[1];
    float* out = (float*)d_out;

    float* ws_s      = (float*)d_ws;                     // 128*300  floats
    float* ws_scores = ws_s + (size_t)BATCH * DFEAT;     // 128*1024 floats

    const int nS = BATCH * DFEAT;
    k0_zero<<<(nS + 255) / 256, 256, 0, stream>>>(ws_s, nS);
    k1_bnorm_accum<<<BATCH * SPLITB, 256, 0, stream>>>(b, ws_s);
    k2_scores_wmma<<<BATCH * SPLITB, 256, 0, stream>>>(a, ws_s, ws_scores);
    k3_softmax<<<BATCH, NROWS, 0, stream>>>(ws_scores);
    k4_tile_out<<<(BATCH * NROWS) / 8, 256, 0, stream>>>(ws_scores, out);
}